// GConv_13537736917293
// MI455X (gfx1250) — compile-verified
//
#include <hip/hip_runtime.h>

// GIN forward: gather -> 3x [scatter-add, GEMM+relu, GEMM+relu, BN] -> concat + pool
// MI455X / gfx1250, wave32. GEMMs use V_WMMA_F32_16X16X4_F32 (fp32, matches ref dtype;
// compute is ~40 GFLOP total and nowhere near the bottleneck -- edge scatter/gather
// L2 traffic dominates, and the whole per-layer working set fits in the 192MB L2).

#define N_NODES 50000
#define N_EDGES 800000
#define DIM_IN  300
#define DIM_H   256
#define N_GRAPH 128
#define VOCAB   11868
#define CAT_W   (3 * DIM_H)   // 768

typedef __attribute__((ext_vector_type(2))) float v2f;
typedef __attribute__((ext_vector_type(8))) float v8f;

#if defined(__has_builtin)
#  if __has_builtin(__builtin_amdgcn_wmma_f32_16x16x4_f32)
#    define HAVE_WMMA_F32X4 1
#  endif
#endif
#ifndef HAVE_WMMA_F32X4
#  define HAVE_WMMA_F32X4 0
#endif

// ---------------------------------------------------------------- utilities
__global__ void zero_kernel(float* __restrict__ p, int n) {
  int i = blockIdx.x * blockDim.x + threadIdx.x;
  if (i < n) p[i] = 0.0f;
}

__global__ void copy4_kernel(const float* __restrict__ src, float* __restrict__ dst, int n4) {
  int i = blockIdx.x * blockDim.x + threadIdx.x;
  if (i < n4) ((float4*)dst)[i] = ((const float4*)src)[i];
}

// ------------------------------------------------------ embedding gather (N x 300)
__global__ void gather_kernel(const int* __restrict__ x, const float* __restrict__ emb,
                              const float* __restrict__ vecr, float* __restrict__ z) {
  const int chunks = DIM_IN / 4;                 // 75 float4 per node
  int idx = blockIdx.x * blockDim.x + threadIdx.x;
  if (idx >= N_NODES * chunks) return;
  int n = idx / chunks, c = idx % chunks;
  int id = x[n];                                  // in [0, VOCAB]; VOCAB -> vec_random
  const float4* sp = (id < VOCAB) ? (const float4*)(emb + (size_t)id * DIM_IN)
                                  : (const float4*)vecr;
  ((float4*)(z + (size_t)n * DIM_IN))[c] = sp[c];
}

// ------------------------------------------------- edge scatter: h[dst] += z[src]
__global__ void edge_scatter_kernel(const float* __restrict__ z, float* __restrict__ h,
                                    const int* __restrict__ src, const int* __restrict__ dst,
                                    int F, int chunks) {
  int idx = blockIdx.x * blockDim.x + threadIdx.x;
  int total = N_EDGES * chunks;                   // <= 60M, fits int
  if (idx >= total) return;
  int e = idx / chunks, c = idx % chunks;         // c fast within wave -> coalesced
  int s = src[e], d = dst[e];
  float4 v = ((const float4*)(z + (size_t)s * F))[c];
  float* hp = h + (size_t)d * F + (size_t)c * 4;
  atomicAdd(hp + 0, v.x);
  atomicAdd(hp + 1, v.y);
  atomicAdd(hp + 2, v.z);
  atomicAdd(hp + 3, v.w);
}

// --------------------------------------- C = relu(A @ B + bias); A: MxK, B: Kx256
// One wave per 16x16 tile; 16 waves/block share one 16-col weight panel in LDS.
__global__ __launch_bounds__(512) void gemm_bias_relu_kernel(
    const float* __restrict__ A, const float* __restrict__ B,
    const float* __restrict__ bias, float* __restrict__ C, int M, int K) {
  __shared__ float Bs[304 * 16];                  // K <= 300 rows of the 16-wide panel

  const int lane   = threadIdx.x & 31;
  const int wave   = threadIdx.x >> 5;
  const int tn     = blockIdx.x & 15;             // 256/16 = 16 column tiles
  const int tmBase = (blockIdx.x >> 4) << 4;      // 16 row-tiles per block group

  // cooperative stage of B[:, tn*16 : tn*16+16] into LDS (conflict-free layout)
  for (int i = threadIdx.x; i < K * 16; i += 512) {
    int k = i >> 4, j = i & 15;
    Bs[i] = B[(size_t)k * DIM_H + tn * 16 + j];
  }
  __syncthreads();

  int tm = tmBase + wave;
  if (tm * 16 >= M) return;                       // tail waves (no further barriers)

  const int nn   = lane & 15;                     // N col within tile / M row for A frag
  const int hi   = lane >> 4;                     // half-wave select
  const int koff = hi << 1;                       // K pair offset per ISA A/B layout
  const int row  = tm * 16 + nn;
  const float* Ap = A + (size_t)row * K + koff;

  v8f acc = {};
#if HAVE_WMMA_F32X4
#pragma unroll 4
  for (int k = 0; k < K; k += 4) {
    v2f a, b;
    a.x = Ap[k];                                  // A[row][k+koff]
    a.y = Ap[k + 1];                              // A[row][k+koff+1]
    b.x = Bs[(k + koff) * 16 + nn];               // B[k+koff][col]
    b.y = Bs[(k + koff + 1) * 16 + nn];           // B[k+koff+1][col]
    acc = __builtin_amdgcn_wmma_f32_16x16x4_f32(
        /*neg_a=*/false, a, /*neg_b=*/false, b,
        /*c_mod=*/(short)0, acc, /*reuse_a=*/false, /*reuse_b=*/false);
  }
#else
  // scalar fallback (diagnostic only; should not trigger on gfx1250)
  for (int r = 0; r < 8; ++r) {
    float s = 0.0f;
    int rw = tm * 16 + r + (hi << 3);
    for (int k = 0; k < K; ++k) s += A[(size_t)rw * K + k] * Bs[k * 16 + nn];
    acc[r] = s;
  }
#endif

  // epilogue: bias + relu + store per ISA C/D layout (VGPR r -> row r + hi*8)
  float bv = bias[tn * 16 + nn];
  int rbase = tm * 16 + (hi << 3);
#pragma unroll
  for (int r = 0; r < 8; ++r) {
    float v = acc[r] + bv;
    C[(size_t)(rbase + r) * DIM_H + tn * 16 + nn] = v > 0.0f ? v : 0.0f;
  }
}

// ------------------------------------------------------- BN statistics (sum, sumsq)
__global__ void bn_stats_kernel(const float* __restrict__ h, float* __restrict__ stats) {
  int c = threadIdx.x;                            // 256 channels, channel-per-lane
  float s = 0.0f, sq = 0.0f;
  for (int n = blockIdx.x; n < N_NODES; n += gridDim.x) {
    float v = h[(size_t)n * DIM_H + c];
    s += v;
    sq += v * v;
  }
  atomicAdd(&stats[c], s);
  atomicAdd(&stats[DIM_H + c], sq);
}

// -------------------- BN apply + write z_cat slice + segment-sum pool into g_cat
__global__ void bn_apply_pool_kernel(const float* __restrict__ h, const float* __restrict__ stats,
                                     const float* __restrict__ gamma, const float* __restrict__ beta,
                                     const int* __restrict__ batch, float* __restrict__ znext,
                                     float* __restrict__ zcat, float* __restrict__ gcat,
                                     int layerOff) {
  int idx = blockIdx.x * blockDim.x + threadIdx.x;
  if (idx >= N_NODES * DIM_H) return;
  int n = idx >> 8, c = idx & 255;
  const float invN = 1.0f / (float)N_NODES;
  float mean = stats[c] * invN;
  float var  = stats[DIM_H + c] * invN - mean * mean;
  float v = (h[idx] - mean) * rsqrtf(var + 1e-5f) * gamma[c] + beta[c];
  znext[idx] = v;
  zcat[(size_t)n * CAT_W + layerOff + c] = v;
  atomicAdd(&gcat[(size_t)batch[n] * CAT_W + layerOff + c], v);
}

// ----------------------------------------------------------------------- driver
extern "C" void kernel_launch(void* const* d_in, const int* in_sizes, int n_in,
                              void* d_out, int out_size, void* d_ws, size_t ws_size,
                              hipStream_t stream) {
  const int*   x     = (const int*)d_in[0];
  const int*   ei    = (const int*)d_in[1];       // (2,E) row-major
  const int*   batch = (const int*)d_in[2];
  const float* emb   = (const float*)d_in[3];
  const float* vecr  = (const float*)d_in[4];
  const float* W1[3]    = {(const float*)d_in[5],  (const float*)d_in[11], (const float*)d_in[17]};
  const float* B1[3]    = {(const float*)d_in[6],  (const float*)d_in[12], (const float*)d_in[18]};
  const float* W2[3]    = {(const float*)d_in[7],  (const float*)d_in[13], (const float*)d_in[19]};
  const float* B2[3]    = {(const float*)d_in[8],  (const float*)d_in[14], (const float*)d_in[20]};
  const float* GAMMA[3] = {(const float*)d_in[9],  (const float*)d_in[15], (const float*)d_in[21]};
  const float* BETA[3]  = {(const float*)d_in[10], (const float*)d_in[16], (const float*)d_in[22]};

  const int* srcIdx = ei;
  const int* dstIdx = ei + N_EDGES;

  float* zcat = (float*)d_out;                    // N x 768
  float* gcat = zcat + (size_t)N_NODES * CAT_W;   // G x 768

  float* zbuf  = (float*)d_ws;                    // N x 300 (layer input)
  float* hbuf  = zbuf + (size_t)N_NODES * DIM_IN; // N x 300 (z + agg / gemm2 out)
  float* tbuf  = hbuf + (size_t)N_NODES * DIM_IN; // N x 256 (gemm1 out)
  float* stats = tbuf + (size_t)N_NODES * DIM_H;  // 512 (sum, sumsq)

  const int TB = 256;

  // z0 = vec_all[x]
  {
    int total = N_NODES * (DIM_IN / 4);
    gather_kernel<<<(total + TB - 1) / TB, TB, 0, stream>>>(x, emb, vecr, zbuf);
  }
  // zero the pooled output (atomics accumulate into it)
  {
    int total = N_GRAPH * CAT_W;
    zero_kernel<<<(total + TB - 1) / TB, TB, 0, stream>>>(gcat, total);
  }

  const int tilesM = N_NODES / 16;                                 // 3125 (exact)
  const int gemmBlocks = ((tilesM + 15) / 16) * 16;                // groups * 16 col-tiles

  for (int l = 0; l < 3; ++l) {
    const int F = (l == 0) ? DIM_IN : DIM_H;
    const int chunks = F / 4;

    // h = z  (so edge atomics produce h = z + agg)
    {
      int n4 = N_NODES * chunks;
      copy4_kernel<<<(n4 + TB - 1) / TB, TB, 0, stream>>>(zbuf, hbuf, n4);
    }
    // h[dst] += z[src] over all edges
    {
      int total = N_EDGES * chunks;
      edge_scatter_kernel<<<(total + TB - 1) / TB, TB, 0, stream>>>(
          zbuf, hbuf, srcIdx, dstIdx, F, chunks);
    }
    // t = relu(h @ W1 + b1)
    gemm_bias_relu_kernel<<<gemmBlocks, 512, 0, stream>>>(hbuf, W1[l], B1[l], tbuf, N_NODES, F);
    // h = relu(t @ W2 + b2)
    gemm_bias_relu_kernel<<<gemmBlocks, 512, 0, stream>>>(tbuf, W2[l], B2[l], hbuf, N_NODES, DIM_H);

    // batchnorm stats
    zero_kernel<<<2, TB, 0, stream>>>(stats, 2 * DIM_H);
    bn_stats_kernel<<<512, DIM_H, 0, stream>>>(hbuf, stats);

    // normalize + emit z_cat slice + pool into g_cat + feed next layer
    {
      int total = N_NODES * DIM_H;
      bn_apply_pool_kernel<<<(total + TB - 1) / TB, TB, 0, stream>>>(
          hbuf, stats, GAMMA[l], BETA[l], batch, zbuf, zcat, gcat, l * DIM_H);
    }
  }
}